// ObjectDetection_57861799412082
// MI455X (gfx1250) — compile-verified
//
#include <hip/hip_runtime.h>
#include <hip/hip_bf16.h>

#define A_N 73728
#define B_N 16
#define M_N 64
#define C_N 80
#define POS_T 0.4f
#define NEG_T 0.1f

#define LOSS_BLOCKS 2048
#define LOSS_THREADS 256
// total float4 elements in logits: B*A*(C/4) = 16*73728*20 = 23,592,960
// = LOSS_BLOCKS*LOSS_THREADS*45 exactly.
#define N4_TOTAL (B_N * A_N * (C_N / 4))
#define LOSS_ITERS 45

typedef __attribute__((ext_vector_type(2))) float v2f;
typedef __attribute__((ext_vector_type(4))) float v4f;
typedef __attribute__((ext_vector_type(8))) float v8f;

// Exact deterministic wave32 sum via V_WMMA_F32_16X16X4_F32.
// A = (x, 0) per lane, B = all ones. D[m][n] = sum_k A[m][k], so
// sum over the 8 D VGPRs gives sum over 8 rows for this column; adding the
// cross-half partner (lane^16, rows 8..15 vs 0..7) gives the full 16-row sum
// = sum of ALL A elements = sum of x over the wave (layout-robust).
__device__ __forceinline__ float wave_sum32(float x) {
  v2f a; a[0] = x;   a[1] = 0.0f;
  v2f b; b[0] = 1.0f; b[1] = 1.0f;
  v8f c = {};
  c = __builtin_amdgcn_wmma_f32_16x16x4_f32(false, a, false, b, (short)0, c,
                                            false, false);
  float s = ((c[0] + c[1]) + (c[2] + c[3])) + ((c[4] + c[5]) + (c[6] + c[7]));
  // SWAPX16 within group-of-32: offset = xor_mask(0x10)<<10 | and_mask(0x1f)
  s += __int_as_float(__builtin_amdgcn_ds_swizzle(__float_as_int(s), 0x401F));
  return s;
}

// ---------------------------------------------------------------- k_prep ---
// Recover integer class from one-hot labels (B*M rows of C floats); zero npos.
__global__ __launch_bounds__(256) void k_prep(const float* __restrict__ labels,
                                              unsigned char* __restrict__ cls,
                                              int* __restrict__ npos) {
  int bm = blockIdx.x * blockDim.x + threadIdx.x;
  if (bm == 0) *npos = 0;
  if (bm < B_N * M_N) {
    const float* row = labels + (size_t)bm * C_N;
    int j = 0;
    for (int c = 0; c < C_N; ++c) {
      if (row[c] > 0.5f) { j = c; break; }
    }
    cls[bm] = (unsigned char)j;
  }
}

// -------------------------------------------------------------- k_assign ---
// Per (batch, anchor): best-IoU argmax over 64 boxes (division-free via
// cross-multiplied comparisons), emit per-anchor code:
//   0..79  -> positive, target class id
//   254    -> negative (fgbg = -1), all-zero targets
//   255    -> ignore   (fgbg = 0), skip entirely
// Boxes staged to LDS with the CDNA5 async global->LDS path.
__global__ __launch_bounds__(256) void k_assign(
    const float4* __restrict__ anchors, const float4* __restrict__ boxes,
    const unsigned char* __restrict__ cls, unsigned char* __restrict__ codes,
    int* __restrict__ npos) {
  __shared__ float4 sbox[M_N];
  __shared__ float sarea[M_N];
  __shared__ unsigned char scls[M_N];
  __shared__ int spos[256];

  const int b = blockIdx.y;
  const int tid = threadIdx.x;

  if (tid < M_N) {
    unsigned lds_off = (unsigned)(size_t)(&sbox[tid]);
    unsigned voff = (unsigned)(tid * 16);
    unsigned long long gbase =
        (unsigned long long)(size_t)(boxes + (size_t)b * M_N);
    asm volatile("global_load_async_to_lds_b128 %0, %1, %2"
                 :
                 : "v"(lds_off), "v"(voff), "s"(gbase)
                 : "memory");
  }
  asm volatile("s_wait_asynccnt 0" ::: "memory");
  __syncthreads();

  if (tid < M_N) {
    float4 bx = sbox[tid];
    sarea[tid] = (bx.z - bx.x) * (bx.w - bx.y);
    scls[tid] = cls[b * M_N + tid];
  }
  __syncthreads();

  const int a_idx = blockIdx.x * 256 + tid;
  const float4 an = anchors[a_idx];
  const float area_a = (an.z - an.x) * (an.w - an.y);

  float bi = 0.0f;   // best inter
  float bu = 1.0f;   // best union (>0 always)
  int bidx = 0;
#pragma unroll 4
  for (int m = 0; m < M_N; ++m) {
    float4 bx = sbox[m];
    float ltx = fmaxf(an.x, bx.x), lty = fmaxf(an.y, bx.y);
    float rbx = fminf(an.z, bx.z), rby = fminf(an.w, bx.w);
    float w = fmaxf(rbx - ltx, 0.0f), h = fmaxf(rby - lty, 0.0f);
    float inter = w * h;
    float uni = area_a + sarea[m] - inter;
    // iou_m > iou_best  <=>  inter*bu > bi*uni   (unions strictly positive)
    bool better = inter * bu > bi * uni;
    bi = better ? inter : bi;
    bu = better ? uni : bu;
    bidx = better ? m : bidx;
  }

  const bool pos = bi > POS_T * bu;  // iou > 0.4
  const bool neg = bi < NEG_T * bu;  // iou < 0.1
  int code = pos ? (int)scls[bidx] : (neg ? 254 : 255);
  codes[(size_t)b * A_N + a_idx] = (unsigned char)code;

  // deterministic integer positive count
  spos[tid] = pos ? 1 : 0;
  __syncthreads();
  for (int s = 128; s > 0; s >>= 1) {
    if (tid < s) spos[tid] += spos[tid + s];
    __syncthreads();
  }
  if (tid == 0 && spos[0] > 0) atomicAdd(npos, spos[0]);
}

// ---------------------------------------------------------------- k_loss ---
// Stream logits as float4 B128 *non-temporal* loads (377 MB read-once > 192 MB
// L2: NT keeps the stream from thrashing L2 so the hot 1.2 MB codes table
// stays resident), skip ignored anchors, accumulate focal-BCE, reduce block
// via WMMA wave-sum + LDS; one partial per block.
__global__ __launch_bounds__(LOSS_THREADS) void k_loss(
    const v4f* __restrict__ lg4, const unsigned char* __restrict__ codes,
    float* __restrict__ partials) {
  const int tid = threadIdx.x;
  const int stride = LOSS_BLOCKS * LOSS_THREADS;
  int i = blockIdx.x * LOSS_THREADS + tid;

  float acc = 0.0f;
#pragma unroll 1
  for (int it = 0; it < LOSS_ITERS; ++it, i += stride) {
    const int anchor = i / 20;              // flat (b*A + a)
    const int c0 = (i - anchor * 20) * 4;   // starting class of this float4
    const int code = codes[anchor];
    if (code != 255) {                      // fgbg != 0
      v4f l4 = __builtin_nontemporal_load(lg4 + i);
#pragma unroll
      for (int k = 0; k < 4; ++k) {
        float l = l4[k];
        float t = (code == c0 + k) ? 1.0f : 0.0f;  // code>=254 never matches
        float e = __expf(-fabsf(l));               // exp(-|l|)
        float lg = __logf(1.0f + e);               // log1p(exp(-|l|))
        float bce = fmaxf(l, 0.0f) - l * t + lg;
        float r = __builtin_amdgcn_rcpf(1.0f + e);
        float p = (l >= 0.0f) ? r : e * r;         // sigmoid(l)
        float w = t - p;
        acc += bce * (w * w);
      }
    }
  }

  float s = wave_sum32(acc);
  __shared__ float swave[LOSS_THREADS / 32];
  if ((tid & 31) == 0) swave[tid >> 5] = s;
  __syncthreads();
  if (tid == 0) {
    float tot = 0.0f;
    for (int w = 0; w < LOSS_THREADS / 32; ++w) tot += swave[w];
    partials[blockIdx.x] = tot;
  }
}

// --------------------------------------------------------------- k_final ---
__global__ __launch_bounds__(256) void k_final(const float* __restrict__ partials,
                                               const int* __restrict__ npos,
                                               float* __restrict__ out) {
  const int tid = threadIdx.x;
  float s = 0.0f;
  for (int i = tid; i < LOSS_BLOCKS; i += 256) s += partials[i];
  float ws = wave_sum32(s);
  __shared__ float sw[8];
  if ((tid & 31) == 0) sw[tid >> 5] = ws;
  __syncthreads();
  if (tid == 0) {
    float tot = 0.0f;
    for (int w = 0; w < 8; ++w) tot += sw[w];
    int np = *npos;
    if (np < 1) np = 1;
    out[0] = tot / (float)np;
  }
}

extern "C" void kernel_launch(void* const* d_in, const int* in_sizes, int n_in,
                              void* d_out, int out_size, void* d_ws,
                              size_t ws_size, hipStream_t stream) {
  (void)in_sizes; (void)n_in; (void)out_size; (void)ws_size;
  const float* logits  = (const float*)d_in[0];  // (B, A, C)
  const float* anchors = (const float*)d_in[1];  // (A, 4)
  const float* boxes   = (const float*)d_in[2];  // (B, M, 4)
  const float* labels  = (const float*)d_in[3];  // (B, M, C)
  float* out = (float*)d_out;

  char* ws = (char*)d_ws;
  int* npos            = (int*)ws;                                  // 4 B
  unsigned char* cls   = (unsigned char*)(ws + 64);                 // 1024 B
  unsigned char* codes = (unsigned char*)(ws + 64 + 1024);          // B*A B
  float* partials = (float*)(ws + 64 + 1024 + (size_t)B_N * A_N);   // 16B-aligned

  k_prep<<<4, 256, 0, stream>>>(labels, cls, npos);
  dim3 g1(A_N / 256, B_N);
  k_assign<<<g1, 256, 0, stream>>>((const float4*)anchors,
                                   (const float4*)boxes, cls, codes, npos);
  k_loss<<<LOSS_BLOCKS, LOSS_THREADS, 0, stream>>>((const v4f*)logits,
                                                   codes, partials);
  k_final<<<1, 256, 0, stream>>>(partials, npos, out);
}